// VOILA_49624052138807
// MI455X (gfx1250) — compile-verified
//
#include <hip/hip_runtime.h>
#include <hip/hip_bf16.h>
#include <stdint.h>

// Problem constants (fixed by setup_inputs)
#define SDIM   96
#define KVOX   (SDIM * SDIM * SDIM)   // 884736 voxels
#define NCLS   118
#define MPAD   128                    // NCLS padded to WMMA tile multiple
#define CFEAT  32
#define NPTS   32768
#define EQCAP  8192                   // capacity for tie-resolution buffer

typedef float v2f __attribute__((ext_vector_type(2)));
typedef float v8f __attribute__((ext_vector_type(8)));

// ---------------------------------------------------------------------------
// Key mapping: kd ascending  <=>  float value descending.
// (standard order-preserving float->uint map, then bitwise NOT)
// ---------------------------------------------------------------------------
__device__ __forceinline__ unsigned int desc_key(float f) {
    unsigned int b = __float_as_uint(f);
    unsigned int asc = (b & 0x80000000u) ? ~b : (b | 0x80000000u);
    return ~asc;
}

// ---------------------------------------------------------------------------
// 1) uncertainty -> descending key + coarse (hi-16-bit) histogram
// ---------------------------------------------------------------------------
__global__ void k_keys_hist(const float* __restrict__ logits,
                            const int*   __restrict__ cls,
                            unsigned int* __restrict__ kd,
                            unsigned int* __restrict__ hist_hi) {
    int k = blockIdx.x * blockDim.x + threadIdx.x;
    if (k >= KVOX) return;
    int c = cls[k];
    float v = logits[(size_t)c * KVOX + k];
    unsigned int key = desc_key(-fabsf(v));
    kd[k] = key;
    atomicAdd(&hist_hi[key >> 16], 1u);
}

// ---------------------------------------------------------------------------
// 2) scan hi histogram: find hi-bin containing the N-th smallest key
//    hdr[2] = hi bin, hdr[3] = count of keys with hi-bin < hdr[2]
// ---------------------------------------------------------------------------
__global__ void k_scan_hi(const unsigned int* __restrict__ hist,
                          unsigned int* __restrict__ hdr) {
    __shared__ unsigned int s[256];
    int t = threadIdx.x;
    unsigned int sum = 0;
    for (int i = 0; i < 256; ++i) sum += hist[t * 256 + i];
    s[t] = sum;
    __syncthreads();
    if (t == 0) {
        unsigned int cum = 0; int g = 0;
        for (; g < 256; ++g) { if (cum + s[g] >= (unsigned)NPTS) break; cum += s[g]; }
        unsigned int base = cum; int b = 0;
        for (; b < 256; ++b) {
            unsigned int h = hist[g * 256 + b];
            if (base + h >= (unsigned)NPTS) break;
            base += h;
        }
        hdr[2] = (unsigned int)(g * 256 + b);
        hdr[3] = base;
    }
}

// ---------------------------------------------------------------------------
// 3) fine (lo-16-bit) histogram restricted to the selected hi-bin
// ---------------------------------------------------------------------------
__global__ void k_hist_lo(const unsigned int* __restrict__ kd,
                          const unsigned int* __restrict__ hdr,
                          unsigned int* __restrict__ hist_lo) {
    int k = blockIdx.x * blockDim.x + threadIdx.x;
    if (k >= KVOX) return;
    unsigned int key = kd[k];
    if ((key >> 16) == hdr[2]) atomicAdd(&hist_lo[key & 0xFFFFu], 1u);
}

// ---------------------------------------------------------------------------
// 4) scan lo histogram: exact threshold key
//    hdr[4]=T, hdr[5]=cntLess(<T), hdr[6]=tieNeed(==T to take, smallest idx)
// ---------------------------------------------------------------------------
__global__ void k_scan_lo(const unsigned int* __restrict__ hist,
                          unsigned int* __restrict__ hdr) {
    __shared__ unsigned int s[256];
    int t = threadIdx.x;
    unsigned int sum = 0;
    for (int i = 0; i < 256; ++i) sum += hist[t * 256 + i];
    s[t] = sum;
    __syncthreads();
    if (t == 0) {
        unsigned int target = (unsigned)NPTS - hdr[3];
        unsigned int cum = 0; int g = 0;
        for (; g < 256; ++g) { if (cum + s[g] >= target) break; cum += s[g]; }
        unsigned int base = cum; int b = 0;
        for (; b < 256; ++b) {
            unsigned int h = hist[g * 256 + b];
            if (base + h >= target) break;
            base += h;
        }
        unsigned int T = (hdr[2] << 16) | (unsigned int)(g * 256 + b);
        hdr[4] = T;
        hdr[5] = hdr[3] + base;
        hdr[6] = (unsigned)NPTS - (hdr[3] + base);
    }
}

// ---------------------------------------------------------------------------
// 5) compact: keys < T -> sel (unordered); keys == T -> eq buffer
// ---------------------------------------------------------------------------
__global__ void k_compact(const unsigned int* __restrict__ kd,
                          unsigned int* __restrict__ hdr,
                          unsigned long long* __restrict__ sel,
                          unsigned int* __restrict__ eq) {
    int k = blockIdx.x * blockDim.x + threadIdx.x;
    if (k >= KVOX) return;
    unsigned int T = hdr[4];
    unsigned int key = kd[k];
    if (key < T) {
        unsigned int p = atomicAdd(&hdr[0], 1u);
        sel[p] = ((unsigned long long)key << 32) | (unsigned int)k;
    } else if (key == T) {
        unsigned int p = atomicAdd(&hdr[1], 1u);
        if (p < EQCAP) eq[p] = (unsigned int)k;   // pathological overflow ignored
    }
}

// ---------------------------------------------------------------------------
// 6) tie resolution: LDS bitonic sort of tied indices ascending, append the
//    tieNeed smallest at sel[cntLess..NPTS)
// ---------------------------------------------------------------------------
__global__ void k_tie(const unsigned int* __restrict__ hdr,
                      const unsigned int* __restrict__ eq,
                      unsigned long long* __restrict__ sel) {
    __shared__ unsigned int s[EQCAP];
    unsigned int E = hdr[1]; if (E > EQCAP) E = EQCAP;
    unsigned int T = hdr[4], cntLess = hdr[5], need = hdr[6];
    int t = threadIdx.x;
    for (int i = t; i < EQCAP; i += blockDim.x)
        s[i] = (i < (int)E) ? eq[i] : 0xFFFFFFFFu;
    __syncthreads();
    for (int kk = 2; kk <= EQCAP; kk <<= 1) {
        for (int j = kk >> 1; j >= 1; j >>= 1) {
            for (int p = t; p < EQCAP / 2; p += blockDim.x) {
                int l = (p % j) + (p / j) * 2 * j;
                int r = l + j;
                bool asc = ((l & kk) == 0);
                unsigned int a = s[l], b = s[r];
                bool sw = asc ? (a > b) : (a < b);
                if (sw) { s[l] = b; s[r] = a; }
            }
            __syncthreads();
        }
    }
    for (int i = t; i < (int)need; i += blockDim.x)
        sel[cntLess + i] = ((unsigned long long)T << 32) | s[i];
}

// ---------------------------------------------------------------------------
// 7) bitonic sort of sel[NPTS] (u64, ascending): global pass (j >= 1024)
// ---------------------------------------------------------------------------
__global__ void k_bitonic_g(unsigned long long* __restrict__ sel, int j, int kk) {
    int i = blockIdx.x * blockDim.x + threadIdx.x;   // i < NPTS (2^15)
    int p = i ^ j;
    if (p > i) {
        bool asc = ((i & kk) == 0);
        unsigned long long a = sel[i], b = sel[p];
        bool sw = asc ? (a > b) : (a < b);
        if (sw) { sel[i] = b; sel[p] = a; }
    }
}

// 7b) LDS pass: handles j = min(512, kk/2) ... 1 over 1024-element spans
__global__ void k_bitonic_l(unsigned long long* __restrict__ sel, int kk) {
    __shared__ unsigned long long s[1024];
    int base = blockIdx.x * 1024;
    int t = threadIdx.x;                              // 512 threads
    s[t] = sel[base + t];
    s[t + 512] = sel[base + t + 512];
    __syncthreads();
    int jstart = ((kk >> 1) < 512) ? (kk >> 1) : 512;
    for (int j = jstart; j >= 1; j >>= 1) {
        int l = (t % j) + (t / j) * 2 * j;
        int gi = base + l;
        bool asc = ((gi & kk) == 0);
        unsigned long long a = s[l], b = s[l + j];
        bool sw = asc ? (a > b) : (a < b);
        if (sw) { s[l] = b; s[l + j] = a; }
        __syncthreads();
    }
    sel[base + t] = s[t];
    sel[base + t + 512] = s[t + 512];
}

// ---------------------------------------------------------------------------
// 8) trilinear point sampling (replicates reference fp32 op order, including
//    the D<->W axis-swap quirk: c0 comes from idx/(W*H) but feeds grid x/W)
//    feats layout: [NPTS][CFEAT] (row per point -> WMMA B fragment loads)
// ---------------------------------------------------------------------------
__global__ void k_sample(const unsigned long long* __restrict__ sel,
                         const float* __restrict__ ff,
                         float* __restrict__ feats) {
    int n = blockIdx.x * blockDim.x + threadIdx.x;
    if (n >= NPTS) return;
    unsigned int idx = (unsigned int)(sel[n] & 0xFFFFFFFFu);
    int di = (int)(idx / (SDIM * SDIM));
    int rem = (int)(idx % (SDIM * SDIM));
    int hi = rem / SDIM;
    int wi = rem % SDIM;
    const float step = 1.0f / (float)SDIM;
    float c0 = 0.5f * step + (float)di * step;
    float c1 = 0.5f * step + (float)hi * step;
    float c2 = 0.5f * step + (float)wi * step;
    float gx = 2.0f * c0 - 1.0f;   // -> W axis
    float gy = 2.0f * c1 - 1.0f;   // -> H axis
    float gz = 2.0f * c2 - 1.0f;   // -> D axis
    float ix = ((gx + 1.0f) * (float)SDIM - 1.0f) * 0.5f;
    float iy = ((gy + 1.0f) * (float)SDIM - 1.0f) * 0.5f;
    float iz = ((gz + 1.0f) * (float)SDIM - 1.0f) * 0.5f;
    float fx0 = floorf(ix), fy0 = floorf(iy), fz0 = floorf(iz);
    float fx = ix - fx0, fy = iy - fy0, fz = iz - fz0;
    int x0 = (int)fx0, y0 = (int)fy0, z0 = (int)fz0;

    float acc[CFEAT];
#pragma unroll
    for (int c = 0; c < CFEAT; ++c) acc[c] = 0.0f;

#pragma unroll
    for (int corner = 0; corner < 8; ++corner) {
        int dx = corner & 1, dy = (corner >> 1) & 1, dz = (corner >> 2) & 1;
        int cx = x0 + dx, cy = y0 + dy, cz = z0 + dz;
        bool valid = (cx >= 0) && (cx < SDIM) && (cy >= 0) && (cy < SDIM) &&
                     (cz >= 0) && (cz < SDIM);
        int ccx = min(max(cx, 0), SDIM - 1);
        int ccy = min(max(cy, 0), SDIM - 1);
        int ccz = min(max(cz, 0), SDIM - 1);
        int lin = (ccz * SDIM + ccy) * SDIM + ccx;
        float w = (dx ? fx : 1.0f - fx) * (dy ? fy : 1.0f - fy) * (dz ? fz : 1.0f - fz);
        w = valid ? w : 0.0f;
        const float* p = ff + lin;
#pragma unroll
        for (int c = 0; c < CFEAT; ++c) acc[c] += w * p[(size_t)c * KVOX];
    }
    float* o = feats + (size_t)n * CFEAT;
#pragma unroll
    for (int c = 0; c < CFEAT; ++c) o[c] = acc[c];
}

// ---------------------------------------------------------------------------
// 8b) pad seg_w [118,32] -> wpad [128,32] (zeros beyond row 117), pad bias.
//     Removes all conditionals from the WMMA fragment loads.
// ---------------------------------------------------------------------------
__global__ void k_pad_w(const float* __restrict__ segw,
                        const float* __restrict__ bias,
                        float* __restrict__ wpad,
                        float* __restrict__ biaspad) {
    int t = threadIdx.x;
    for (int i = t; i < MPAD * CFEAT; i += blockDim.x) {
        int m = i / CFEAT;
        wpad[i] = (m < NCLS) ? segw[i] : 0.0f;
    }
    for (int m = t; m < MPAD; m += blockDim.x)
        biaspad[m] = (m < NCLS) ? bias[m] : 0.0f;
}

// ---------------------------------------------------------------------------
// 9) classifier GEMM via V_WMMA_F32_16X16X4_F32:
//    out[m,n] = sum_k wpad[m,k] * feats[n,k] + bias[m]
//    One wave -> one 16x16 output tile, 8 chained WMMAs over K=32.
//    All fragment loads are unconditional aligned b64 loads issued before the
//    WMMA chain; EXEC stays all-ones through the WMMA region.
//    A 16x4 layout: lane(M=lane&15), K = (lane>>4)*2 + {vgpr0,vgpr1}
//    B 4x16 layout: lane(N=lane&15), K = (lane>>4)*2 + {vgpr0,vgpr1}
//    C/D: vgpr r -> M = (lane>>4)*8 + r, N = lane&15
// ---------------------------------------------------------------------------
__global__ void k_gemm(const float* __restrict__ feats,
                       const float* __restrict__ wpad,
                       const float* __restrict__ biaspad,
                       float* __restrict__ out) {
    int wid  = blockIdx.x * (blockDim.x >> 5) + (threadIdx.x >> 5);
    int lane = threadIdx.x & 31;
    int mtile = wid & 7;      // 8 tiles cover 128 padded rows (118 used)
    int ntile = wid >> 3;     // 2048 tiles cover NPTS columns
    int m  = mtile * 16 + (lane & 15);
    int n  = ntile * 16 + (lane & 15);
    int kb = (lane >> 4) * 2;

    const float* ap = wpad + m * CFEAT + kb;               // 8B aligned (kb even)
    const float* bp = feats + (size_t)n * CFEAT + kb;      // 8B aligned

    v2f a[8], b[8];
#pragma unroll
    for (int kk = 0; kk < 8; ++kk) {
        a[kk] = *(const v2f*)(ap + kk * 4);
        b[kk] = *(const v2f*)(bp + kk * 4);
    }

    v8f acc = {};
#pragma unroll
    for (int kk = 0; kk < 8; ++kk)
        acc = __builtin_amdgcn_wmma_f32_16x16x4_f32(
            /*neg_a=*/false, a[kk], /*neg_b=*/false, b[kk],
            /*c_mod=*/(short)0, acc, /*reuse_a=*/false, /*reuse_b=*/false);

    int mbase = mtile * 16 + ((lane >> 4) * 8);
#pragma unroll
    for (int r = 0; r < 8; ++r) {
        int mm = mbase + r;
        if (mm < NCLS) out[(size_t)mm * NPTS + n] = acc[r] + biaspad[mm];
    }
}

// ---------------------------------------------------------------------------
// Host launcher
// ---------------------------------------------------------------------------
extern "C" void kernel_launch(void* const* d_in, const int* in_sizes, int n_in,
                              void* d_out, int out_size, void* d_ws, size_t ws_size,
                              hipStream_t stream) {
    const float* logits = (const float*)d_in[0];   // [1,118,96,96,96]
    const float* ff     = (const float*)d_in[1];   // [1,32,96,96,96]
    const float* segw   = (const float*)d_in[2];   // [118,32]
    const float* bias   = (const float*)d_in[3];   // [118]
    const int*   cls    = (const int*)d_in[4];     // [1,96,96,96]
    float* out = (float*)d_out;                    // [118, 32768]
    (void)in_sizes; (void)n_in; (void)out_size; (void)ws_size;

    // workspace layout (bytes)
    char* ws = (char*)d_ws;
    unsigned int* hdr      = (unsigned int*)(ws + 0);           // 64 B
    unsigned int* hist_hi  = (unsigned int*)(ws + 64);          // 256 KB
    unsigned int* hist_lo  = (unsigned int*)(ws + 64 + 262144); // 256 KB
    unsigned int* kd       = (unsigned int*)(ws + 64 + 2 * 262144);
    size_t off = 64 + 2 * 262144 + (size_t)KVOX * 4;            // 4,063,296 (8B aligned)
    unsigned long long* sel = (unsigned long long*)(ws + off);
    off += (size_t)NPTS * 8;
    unsigned int* eq = (unsigned int*)(ws + off);
    off += (size_t)EQCAP * 4;
    float* feats = (float*)(ws + off);                          // [NPTS][CFEAT]
    off += (size_t)NPTS * CFEAT * 4;
    float* wpad = (float*)(ws + off);                           // [128][32]
    off += (size_t)MPAD * CFEAT * 4;
    float* biaspad = (float*)(ws + off);                        // [128]

    // zero header + both histograms (graph-capturable memset)
    hipMemsetAsync(d_ws, 0, 64 + 2 * 262144, stream);

    const int TB = 256;
    const int gK = (KVOX + TB - 1) / TB;

    k_pad_w    <<<1, 256, 0, stream>>>(segw, bias, wpad, biaspad);
    k_keys_hist<<<gK, TB, 0, stream>>>(logits, cls, kd, hist_hi);
    k_scan_hi  <<<1, 256, 0, stream>>>(hist_hi, hdr);
    k_hist_lo  <<<gK, TB, 0, stream>>>(kd, hdr, hist_lo);
    k_scan_lo  <<<1, 256, 0, stream>>>(hist_lo, hdr);
    k_compact  <<<gK, TB, 0, stream>>>(kd, hdr, sel, eq);
    k_tie      <<<1, 256, 0, stream>>>(hdr, eq, sel);

    // full bitonic sort of 32768 (key=(kd<<32)|idx, ascending)
    for (int kk = 2; kk <= NPTS; kk <<= 1) {
        for (int j = kk >> 1; j >= 1024; j >>= 1)
            k_bitonic_g<<<NPTS / TB, TB, 0, stream>>>(sel, j, kk);
        k_bitonic_l<<<NPTS / 1024, 512, 0, stream>>>(sel, kk);
    }

    k_sample<<<NPTS / TB, TB, 0, stream>>>(sel, ff, feats);

    // 16384 waves (8 m-tiles x 2048 n-tiles), 8 waves/block
    k_gemm<<<2048, 256, 0, stream>>>(feats, wpad, biaspad, out);
}